// CG_woFilter_cuda_3813930959610
// MI455X (gfx1250) — compile-verified
//
#include <hip/hip_runtime.h>
#include <math.h>

typedef __attribute__((ext_vector_type(2))) float v2f;
typedef __attribute__((ext_vector_type(8))) float v8f;

#define LMAX     5
#define NTRIPLES 69
#define NMSLICES 457
#define IN_ROWS  576
#define OUT_ROWS 116992
#define BATCH    256

// d_ws layout (int granularity)
#define TRIPLE_OFF_I 0     // 69 * 8 ints  {l,l1,l2,f1Base,f2Base,outRow,msBegin,nm}
#define MSLICE_OFF_I 1024  // 457 * 2 ints {slotBegin, nChunks}   (byte 4096)
#define SLOT_OFF_I   2048  // slots: 2 ints each {f32 coeff, meta} (byte 8192)

__device__ __constant__ int d_CUM[7] = {0, 16, 64, 144, 256, 400, 576};

__device__ __forceinline__ double dfact_tab(const double* f, int n) { return f[n]; }

__device__ double cgcoef(const double* f, int l1, int m1, int l2, int m2, int l, int m) {
  if (m1 + m2 != m) return 0.0;
  double pre = sqrt((double)(2 * l + 1) * f[l + l1 - l2] * f[l - l1 + l2] *
                    f[l1 + l2 - l] / f[l1 + l2 + l + 1]);
  pre *= sqrt(f[l + m] * f[l - m] * f[l1 - m1] * f[l1 + m1] * f[l2 - m2] * f[l2 + m2]);
  int kmin = max(0, max(l2 - l - m1, l1 + m2 - l));
  int kmax = min(l1 + l2 - l, min(l1 - m1, l2 + m2));
  double s = 0.0;
  for (int k = kmin; k <= kmax; ++k) {
    double d = f[k] * f[l1 + l2 - l - k] * f[l1 - m1 - k] * f[l2 + m2 - k] *
               f[l - l2 + m1 + k] * f[l - l1 - m2 + k];
    s += ((k & 1) ? -1.0 : 1.0) / d;
  }
  return pre * s;
}

// Regenerates all CG metadata into d_ws (deterministic; runs every call).
// One thread per (triple, m) slice; integer-only structural scan per thread.
__global__ void cg_prep(int* __restrict__ ws) {
  const int tid = blockIdx.x * blockDim.x + threadIdx.x;
  if (tid >= NMSLICES) return;
  double f[16];
  f[0] = 1.0;
  for (int k = 1; k < 16; ++k) f[k] = f[k - 1] * (double)k;

  int msIdx = 0, slotIdx = 0, outRow = 0, t = 0;
  for (int l = 0; l <= LMAX; ++l)
    for (int l1 = 0; l1 <= LMAX; ++l1)
      for (int l2 = 0; l2 <= l1; ++l2) {
        if (!(abs(l1 - l2) <= l && l <= l1 + l2)) continue;
        const int nm = 2 * l + 1;
        if (tid == msIdx) {  // first m-slice thread of this triple writes header
          int* tp = ws + TRIPLE_OFF_I + t * 8;
          tp[0] = l; tp[1] = l1; tp[2] = l2;
          tp[3] = d_CUM[l1]; tp[4] = d_CUM[l2];
          tp[5] = outRow; tp[6] = msIdx; tp[7] = nm;
        }
        for (int mm = 0; mm < nm; ++mm) {
          const int mmm = mm - l;
          const int p_lo = max(-l1, mmm - l2), p_hi = min(l1, mmm + l2);
          const int K = p_hi - p_lo + 1;          // valid (m1,m2) pairs (zeros kept)
          const int ns = ((2 * K + 3) >> 2) << 2; // complex-stacked, padded to 4
          if (tid == msIdx) {
            int* msp = ws + MSLICE_OFF_I + msIdx * 2;
            msp[0] = slotIdx; msp[1] = ns >> 2;
            float*    sc = (float*)(ws + SLOT_OFF_I) ;
            unsigned* sm = (unsigned*)(ws + SLOT_OFF_I);
            int w = slotIdx;
            for (int mm1 = p_lo; mm1 <= p_hi; ++mm1) {
              const int mm2 = mmm - mm1;
              const float c = (float)cgcoef(f, l1, mm1, l2, mm2, l, mmm);
              const unsigned meta = (unsigned)(mm1 + l1) | ((unsigned)(mm2 + l2) << 8);
              sc[w * 2] = c; sm[w * 2 + 1] = meta;              ++w;  // real slot
              sc[w * 2] = c; sm[w * 2 + 1] = meta | (1u << 16); ++w;  // imag slot
            }
            for (; w < slotIdx + ns; ++w) { sc[w * 2] = 0.f; sm[w * 2 + 1] = 0u; }
          }
          msIdx++; slotIdx += ns;
        }
        outRow += 256 * nm;
        t++;
      }
}

// One wave per (triple, batch element). Each (m) slice: 16x16 f32 tile built by
// V_WMMA_F32_16X16X4_F32 accumulation over complex-stacked K slots.
__global__ __launch_bounds__(256) void cg_wmma(const float* __restrict__ act,
                                               float* __restrict__ out,
                                               const int* __restrict__ ws) {
  const int lane = threadIdx.x & 31;
  const int wave = threadIdx.x >> 5;
  const int t = blockIdx.x;
  const int b = blockIdx.y * 8 + wave;

  const int* tp = ws + TRIPLE_OFF_I + t * 8;
  const int l1 = tp[1], l2 = tp[2];
  const int f1Base = tp[3], f2Base = tp[4];
  const int outRow = tp[5], msBegin = tp[6], nm = tp[7];

  const int li = lane & 15;  // A row (i) / B col (j) / D col (N)
  const int hi = lane >> 4;  // K half-block select; D row block select

  const float* f1p = act + (size_t)((b * IN_ROWS + f1Base + li * (2 * l1 + 1)) * 2);
  const float* f2p = act + (size_t)((b * IN_ROWS + f2Base + li * (2 * l2 + 1)) * 2);
  float* outp = out + ((size_t)b * OUT_ROWS + outRow) * 2;
  const unsigned long long* __restrict__ slots =
      (const unsigned long long*)(ws + SLOT_OFF_I);

  for (int m = 0; m < nm; ++m) {
    const int* msp = ws + MSLICE_OFF_I + (msBegin + m) * 2;
    int sbase = msp[0];
    const int nchunks = msp[1];
    v8f accR = {};
    v8f accI = {};
    for (int c = 0; c < nchunks; ++c) {
      // 4 slot records per chunk: uniform (scalar) loads, lane-half select.
      const unsigned long long s0 = slots[sbase + 0];
      const unsigned long long s1 = slots[sbase + 1];
      const unsigned long long s2 = slots[sbase + 2];
      const unsigned long long s3 = slots[sbase + 3];
      sbase += 4;
      const unsigned long long sa = hi ? s2 : s0;  // K slot = 2*hi + 0
      const unsigned long long sb = hi ? s3 : s1;  // K slot = 2*hi + 1
      v2f aR, aI, bR, bI;
#pragma unroll
      for (int kk = 0; kk < 2; ++kk) {
        const unsigned long long s = kk ? sb : sa;
        const float cw = __builtin_bit_cast(float, (unsigned)(s & 0xffffffffu));
        const unsigned meta = (unsigned)(s >> 32);
        const int m1 = meta & 255;
        const int m2 = (meta >> 8) & 255;
        const bool fl = (meta >> 16) & 1;  // 0: real slot, 1: imag slot
        const v2f f1 = *(const v2f*)(f1p + 2 * m1);
        const v2f f2 = *(const v2f*)(f2p + 2 * m2);
        const float f1v = fl ? f1.y : f1.x;
        const float av = cw * f1v;
        aR[kk] = fl ? -av : av;      // chain R: +C*F1r / -C*F1i
        aI[kk] = av;                 // chain I: +C*F1r / +C*F1i
        bR[kk] = fl ? f2.y : f2.x;   // chain R: F2r / F2i
        bI[kk] = fl ? f2.x : f2.y;   // chain I: F2i / F2r
      }
      accR = __builtin_amdgcn_wmma_f32_16x16x4_f32(false, aR, false, bR,
                                                   (short)0, accR, false, false);
      accI = __builtin_amdgcn_wmma_f32_16x16x4_f32(false, aI, false, bI,
                                                   (short)0, accI, false, false);
    }
    // D layout: lane 0-15 -> M=r, lane 16-31 -> M=8+r; N = lane&15.
    // (m, re/im) is innermost in memory -> contiguous 8B NT stores.
#pragma unroll
    for (int r = 0; r < 8; ++r) {
      const int rowM = hi * 8 + r;
      const size_t idx = ((size_t)(rowM * 16 + li) * nm + m) * 2;
      v2f v;
      v.x = accR[r];
      v.y = accI[r];
      __builtin_nontemporal_store(v, (v2f*)(outp + idx));
    }
  }
}

extern "C" void kernel_launch(void* const* d_in, const int* in_sizes, int n_in,
                              void* d_out, int out_size, void* d_ws, size_t ws_size,
                              hipStream_t stream) {
  const float* act = (const float*)d_in[0];
  float* out = (float*)d_out;
  int* ws = (int*)d_ws;
  (void)in_sizes; (void)n_in; (void)out_size; (void)ws_size;

  hipLaunchKernelGGL(cg_prep, dim3(2), dim3(256), 0, stream, ws);
  dim3 grid(NTRIPLES, BATCH / 8);
  hipLaunchKernelGGL(cg_wmma, grid, dim3(256), 0, stream, act, out, ws);
}